// GCN_68736656605911
// MI455X (gfx1250) — compile-verified
//
#include <hip/hip_runtime.h>
#include <hip/hip_bf16.h>

typedef float v2f __attribute__((ext_vector_type(2)));
typedef float v8f __attribute__((ext_vector_type(8)));
typedef unsigned int u32;
typedef u32 u32x4 __attribute__((ext_vector_type(4)));
typedef u32 u32x8 __attribute__((ext_vector_type(8)));

#define NFEAT 512
#define NHID  128
#define NCLS  40

// LDS offset (bytes) of a __shared__ object: generic -> addrspace(3) -> int.
__device__ __forceinline__ u32 lds_addr_of(const void* p) {
  return (u32)(unsigned long long)(__attribute__((address_space(3))) const char*)p;
}

// ---------------------------------------------------------------------------
// GEMM1: S1[n,128] = X[n,512] @ W1[512,128], fp32 WMMA 16x16x4.
// Block = 256 threads (8 waves); block owns one 16-row tile, wave w owns cols
// [16w,16w+16). The 16x512 x-tile is staged into LDS by the Tensor Data Mover
// (tensor_load_to_lds, TENSORcnt), with TDM padding of 1 dword per 128 dwords
// -> effective row stride 516 floats, 128-dword block stride 129 floats.
// Bank map for A reads: (m*516 + C) % 64 = m*4 + C' -> all 32 lanes distinct.
// ---------------------------------------------------------------------------
__global__ __launch_bounds__(256) void gcn_gemm1(const float* __restrict__ X,
                                                 const float* __restrict__ W1,
                                                 float* __restrict__ S1, int n) {
  __shared__ float xs[16 * 516];
  const int tid  = threadIdx.x;
  const int r0   = blockIdx.x * 16;
  const bool full = (r0 + 16 <= n);

  if (full) {
    if (tid < 32) {  // wave 0 issues the TDM descriptor (EXEC ignored by TDM)
      const unsigned long long ga = (unsigned long long)(X + (size_t)r0 * NFEAT);
      const u32 rows = (u32)(n - r0);
      u32x4 g0;
      g0.x = 1u;                                        // count=1, no gather
      g0.y = lds_addr_of(xs);                           // lds_addr (bytes)
      g0.z = (u32)ga;                                   // global_addr[31:0]
      g0.w = (u32)((ga >> 32) & 0x1FFFFFFu) | (2u << 30); // ga[56:32] | type=2
      u32x8 g1;
      g1.s0 = (2u << 16)        // data_size = 4B
            | (1u << 20)        // pad_enable
            | (6u << 22)        // pad_interval: every 128 dwords
            | (0u << 25);       // pad_amount: 1 dword
      g1.s1 = (512u & 0xFFFFu) << 16;                     // tensor_dim0[15:0]
      g1.s2 = 0u | ((rows & 0xFFFFu) << 16);              // td0[31:16] | td1[15:0]
      g1.s3 = ((rows >> 16) & 0xFFFFu) | (512u << 16);    // td1[31:16] | tile_dim0
      g1.s4 = 16u;                                        // tile_dim1=16, tile_dim2=0
      g1.s5 = 512u;                                       // tensor_dim0_stride[31:0]
      g1.s6 = 0u;
      g1.s7 = 0u;
      asm volatile("tensor_load_to_lds %0, %1" :: "s"(g0), "s"(g1) : "memory");
      __builtin_amdgcn_s_wait_tensorcnt(0);
    }
  } else {
    // tail tile: cooperative load into the identical padded layout
    for (int idx = tid; idx < 16 * 128; idx += 256) {
      const int row = idx >> 7;
      const int c4  = (idx & 127) << 2;
      float4 v = make_float4(0.f, 0.f, 0.f, 0.f);
      if (r0 + row < n) v = *(const float4*)&X[(size_t)(r0 + row) * NFEAT + c4];
      const int o = row * 516 + c4 + (c4 >> 7);
      xs[o + 0] = v.x; xs[o + 1] = v.y; xs[o + 2] = v.z; xs[o + 3] = v.w;
    }
  }
  __syncthreads();

  const int lane = tid & 31;
  const int wave = tid >> 5;            // 0..7
  const int c0   = wave * 16;
  const int m    = lane & 15;
  const int kk   = (lane < 16) ? 0 : 2; // K sub-offset per lane half
  const int c    = c0 + m;

  v8f acc = {};
  const float* __restrict__ ap = &xs[m * 516 + kk];
  const float* __restrict__ bp = &W1[kk * NHID + c];
#pragma unroll 1
  for (int kb = 0; kb < 4; ++kb) {
    const float* a0 = ap + kb * 129;        // 128-dword block + 1-dword TDM pad
    const float* b0 = bp + kb * 128 * NHID;
#pragma unroll 8
    for (int k2 = 0; k2 < 128; k2 += 4) {
      v2f a;
      a.x = a0[k2];
      a.y = a0[k2 + 1];
      v2f b;
      b.x = b0[(k2)     * NHID];
      b.y = b0[(k2 + 1) * NHID];
      acc = __builtin_amdgcn_wmma_f32_16x16x4_f32(false, a, false, b,
                                                  (short)0, acc, false, false);
    }
  }

  // D layout: VGPR r -> M=r (lanes 0-15) / M=8+r (lanes 16-31), N=lane%16
  const int rbase = r0 + ((lane < 16) ? 0 : 8);
  float* __restrict__ op = &S1[(size_t)rbase * NHID + c];
  if (full) {
#pragma unroll
    for (int r = 0; r < 8; ++r) op[r * NHID] = acc[r];
  } else {
#pragma unroll
    for (int r = 0; r < 8; ++r)
      if (rbase + r < n) op[r * NHID] = acc[r];
  }
}

// ---------------------------------------------------------------------------
// SpMM layer 1: H[row[e], :] += w[e] * S1[col[e], :]   (128 feats, L2-resident)
// ---------------------------------------------------------------------------
__global__ __launch_bounds__(128) void gcn_spmm1(const int* __restrict__ er,
                                                 const int* __restrict__ ec,
                                                 const float* __restrict__ ew,
                                                 const float* __restrict__ S1,
                                                 float* __restrict__ H, int e) {
  const int j    = threadIdx.x;
  const int base = blockIdx.x * 16;
  const int lim  = (base + 16 <= e) ? 16 : (e - base);
#pragma unroll 1
  for (int i = 0; i < lim; ++i) {
    const int eid = base + i;
    const int   r = er[eid];
    const int   c = ec[eid];
    const float w = ew[eid];
    atomicAdd(&H[(size_t)r * NHID + j], w * S1[(size_t)c * NHID + j]);
  }
}

// ---------------------------------------------------------------------------
// Epilogue 1: x2 = relu(H + b1) * drop_mask  -> written straight to output #2
// ---------------------------------------------------------------------------
__global__ __launch_bounds__(256) void gcn_epi1(const float* __restrict__ H,
                                                const float* __restrict__ b1,
                                                const float* __restrict__ mask,
                                                float* __restrict__ X2, int total) {
  const int t = blockIdx.x * 256 + threadIdx.x;
  if (t >= total) return;
  const int j = t & (NHID - 1);
  float v = H[t] + b1[j];
  v = v > 0.f ? v : 0.f;
  X2[t] = v * mask[t];
}

// ---------------------------------------------------------------------------
// GEMM2: S2[n,40] = X2[n,128] @ W2[128,40], cols padded 40->48 (3 col tiles).
// Block = 96 threads (3 waves), one 16-row tile, wave w owns cols [16w,16w+16).
// ---------------------------------------------------------------------------
__global__ __launch_bounds__(96) void gcn_gemm2(const float* __restrict__ X2,
                                                const float* __restrict__ W2,
                                                float* __restrict__ S2, int n) {
  __shared__ float xs[16 * 132];
  const int tid  = threadIdx.x;
  const int r0   = blockIdx.x * 16;
  const bool full = (r0 + 16 <= n);

  for (int idx = tid; idx < 16 * 32; idx += 96) {
    const int row = idx >> 5;
    const int c4  = (idx & 31) << 2;
    float4 v = make_float4(0.f, 0.f, 0.f, 0.f);
    if (r0 + row < n) v = *(const float4*)&X2[(size_t)(r0 + row) * NHID + c4];
    xs[row * 132 + c4 + 0] = v.x;
    xs[row * 132 + c4 + 1] = v.y;
    xs[row * 132 + c4 + 2] = v.z;
    xs[row * 132 + c4 + 3] = v.w;
  }
  __syncthreads();

  const int lane = tid & 31;
  const int wave = tid >> 5;            // 0..2
  const int c0   = wave * 16;
  const int m    = lane & 15;
  const int kk   = (lane < 16) ? 0 : 2;
  const int c    = c0 + m;
  const bool cok = c < NCLS;

  v8f acc = {};
#pragma unroll 8
  for (int k = 0; k < NHID; k += 4) {
    v2f a = *(const v2f*)&xs[m * 132 + k + kk];
    v2f b;
    b.x = cok ? W2[(k + kk)     * NCLS + c] : 0.f;
    b.y = cok ? W2[(k + kk + 1) * NCLS + c] : 0.f;
    acc = __builtin_amdgcn_wmma_f32_16x16x4_f32(false, a, false, b,
                                                (short)0, acc, false, false);
  }

  const int rbase = r0 + ((lane < 16) ? 0 : 8);
  float* __restrict__ op = &S2[(size_t)rbase * NCLS + c];
  if (full) {
    if (cok) {
#pragma unroll
      for (int r = 0; r < 8; ++r) op[r * NCLS] = acc[r];
    }
  } else {
#pragma unroll
    for (int r = 0; r < 8; ++r)
      if (cok && rbase + r < n) op[r * NCLS] = acc[r];
  }
}

// ---------------------------------------------------------------------------
// SpMM layer 2: X3[row[e], f] += w[e] * S2[col[e], f], 64 threads/edge, f<40
// ---------------------------------------------------------------------------
__global__ __launch_bounds__(256) void gcn_spmm2(const int* __restrict__ er,
                                                 const int* __restrict__ ec,
                                                 const float* __restrict__ ew,
                                                 const float* __restrict__ S2,
                                                 float* __restrict__ X3, int e) {
  const int t   = blockIdx.x * 256 + threadIdx.x;
  const int eid = t >> 6;
  const int f   = t & 63;
  if (eid >= e || f >= NCLS) return;
  const int   r = er[eid];
  const int   c = ec[eid];
  const float w = ew[eid];
  atomicAdd(&X3[(size_t)r * NCLS + f], w * S2[(size_t)c * NCLS + f]);
}

// ---------------------------------------------------------------------------
// log_softmax over 40 classes: one wave32 per row, shfl_xor reductions.
// ---------------------------------------------------------------------------
__global__ __launch_bounds__(256) void gcn_lsm(const float* __restrict__ X3,
                                               const float* __restrict__ b2,
                                               float* __restrict__ out, int n) {
  const int r    = blockIdx.x * 8 + (threadIdx.x >> 5);
  const int lane = threadIdx.x & 31;
  if (r >= n) return;
  const float NEG_INF = -__builtin_inff();

  const float v0 = X3[(size_t)r * NCLS + lane] + b2[lane];          // lanes 0..31
  const float v1 = (lane < 8)
                       ? X3[(size_t)r * NCLS + 32 + lane] + b2[32 + lane]
                       : NEG_INF;

  float mx = fmaxf(v0, v1);
#pragma unroll
  for (int off = 16; off > 0; off >>= 1)
    mx = fmaxf(mx, __shfl_xor(mx, off, 32));

  float s = __expf(v0 - mx) + ((lane < 8) ? __expf(v1 - mx) : 0.f);
#pragma unroll
  for (int off = 16; off > 0; off >>= 1)
    s += __shfl_xor(s, off, 32);

  const float lse = mx + __logf(s);
  out[(size_t)r * NCLS + lane] = v0 - lse;
  if (lane < 8) out[(size_t)r * NCLS + 32 + lane] = v1 - lse;
}

// ---------------------------------------------------------------------------
extern "C" void kernel_launch(void* const* d_in, const int* in_sizes, int n_in,
                              void* d_out, int out_size, void* d_ws, size_t ws_size,
                              hipStream_t stream) {
  const float* x    = (const float*)d_in[0];
  const int*   erow = (const int*)d_in[1];
  const int*   ecol = (const int*)d_in[2];
  const float* ew   = (const float*)d_in[3];
  const float* W1   = (const float*)d_in[4];
  const float* b1   = (const float*)d_in[5];
  const float* W2   = (const float*)d_in[6];
  const float* b2   = (const float*)d_in[7];
  const float* mask = (const float*)d_in[8];

  const int n = in_sizes[0] / NFEAT;  // 100000
  const int e = in_sizes[1];          // 3200000

  float* out1 = (float*)d_out;                      // [n,40] log_softmax
  float* out2 = (float*)d_out + (size_t)n * NCLS;   // [n,128] x2

  float* S1 = (float*)d_ws;                         // [n,128]
  float* H  = S1 + (size_t)n * NHID;                // [n,128] accumulator
  float* S2 = H  + (size_t)n * NHID;                // [n,40]
  float* X3 = S2 + (size_t)n * NCLS;                // [n,40]  accumulator

  hipMemsetAsync(H,  0, (size_t)n * NHID * sizeof(float), stream);
  hipMemsetAsync(X3, 0, (size_t)n * NCLS * sizeof(float), stream);

  gcn_gemm1<<<(n + 15) / 16, 256, 0, stream>>>(x, W1, S1, n);
  gcn_spmm1<<<(e + 15) / 16, 128, 0, stream>>>(erow, ecol, ew, S1, H, e);
  gcn_epi1 <<<(n * NHID + 255) / 256, 256, 0, stream>>>(H, b1, mask, out2, n * NHID);
  gcn_gemm2<<<(n + 15) / 16, 96, 0, stream>>>(out2, W2, S2, n);
  gcn_spmm2<<<(e + 3) / 4, 256, 0, stream>>>(erow, ecol, ew, S2, X3, e);
  gcn_lsm  <<<(n + 7) / 8, 256, 0, stream>>>(X3, b2, out1, n);
}